// LConvBilin_43800076485325
// MI455X (gfx1250) — compile-verified
//
#include <hip/hip_runtime.h>

#define BX   16
#define CCH  2
#define XD   512
#define YD   512
#define NOUT 8

typedef float v2f __attribute__((ext_vector_type(2)));
typedef float v8f __attribute__((ext_vector_type(8)));

__device__ __forceinline__ float2 mkc(float x, float y) { float2 r; r.x = x; r.y = y; return r; }
__device__ __forceinline__ float2 cadd(float2 a, float2 b) { return mkc(a.x + b.x, a.y + b.y); }
__device__ __forceinline__ float2 csub(float2 a, float2 b) { return mkc(a.x - b.x, a.y - b.y); }
__device__ __forceinline__ float2 cI(float2 a)             { return mkc(-a.y, a.x); }           // i*z
__device__ __forceinline__ float2 cmulc(float2 a, float2 b){ return mkc(a.x*b.x - a.y*b.y, a.x*b.y + a.y*b.x); }

// Out[M=2i+t, N=pixel] = sum_K Acoef[M,K] * feat[K, pixel]   (K = 68, 17 x WMMA f32 16x16x4)
__global__ __launch_bounds__(256) void lconv_wmma_kernel(
    const float2* __restrict__ plaqs,   // (B, 2, X, Y) complex
    const float2* __restrict__ links,   // (B, 2, X, Y) complex, unit modulus
    const float2* __restrict__ alpha,   // (8, 5, 13) complex
    float2* __restrict__ out)           // (B, 8, X, Y) complex
{
  __shared__ float Bmat[68 * 16];       // Bmat[K*16 + (2i+t)] = component of complex coeff c_{K,i}
  const int tid = threadIdx.x;

  // ---- build alpha-derived coefficient matrix (544 complex coeffs) ----
  for (int task = tid; task < 544; task += 256) {
    const int K = task >> 3;
    const int i = task & 7;
    float2 z;
    if (K < 48) {
      // product features: K = 4*(c*6+k) + q, q in {RR, RI, IR, II}
      const int s = K >> 2, q = K & 3;
      const int c = s / 6, k = s % 6;
      const float2 A1 = alpha[(i*5 + c    )*13 + k    ];  // W_c * T_k
      const float2 A2 = alpha[(i*5 + c    )*13 + k + 6];  // W_c * conj(T_k)
      const float2 A3 = alpha[(i*5 + c + 2)*13 + k    ];  // conj(W_c) * T_k
      const float2 A4 = alpha[(i*5 + c + 2)*13 + k + 6];  // conj(W_c) * conj(T_k)
      if      (q == 0) z = cadd(cadd(A1, A2), cadd(A3, A4));          // ReW*ReT
      else if (q == 1) z = cI(cadd(csub(A1, A2), csub(A3, A4)));      // ReW*ImT
      else if (q == 2) z = cI(csub(cadd(A1, A2), cadd(A3, A4)));      // ImW*ReT
      else             z = csub(cadd(A2, A3), cadd(A1, A4));          // ImW*ImT
    } else if (K < 52) {
      // linear-in-W features: K=48..51 -> ReW0, ImW0, ReW1, ImW1
      const int c = (K - 48) >> 1, im = (K - 48) & 1;
      const float2 A  = alpha[(i*5 + c    )*13 + 12];
      const float2 Bv = alpha[(i*5 + c + 2)*13 + 12];
      z = im ? cI(csub(A, Bv)) : cadd(A, Bv);
    } else if (K < 64) {
      // linear-in-T features: K=52..63 -> ReT_k, ImT_k
      const int k = (K - 52) >> 1, im = (K - 52) & 1;
      const float2 A  = alpha[(i*5 + 4)*13 + k    ];
      const float2 Bv = alpha[(i*5 + 4)*13 + k + 6];
      z = im ? cI(csub(A, Bv)) : cadd(A, Bv);
    } else if (K == 64) {
      z = alpha[(i*5 + 4)*13 + 12];   // constant term
    } else {
      z = mkc(0.f, 0.f);              // K padding 65..67
    }
    Bmat[K*16 + 2*i    ] = z.x;
    Bmat[K*16 + 2*i + 1] = z.y;
  }
  __syncthreads();

  const int lane = tid & 31;
  const int wave = tid >> 5;
  const int l = lane & 15;     // A: output-component row M; B: pixel column N
  const int h = lane >> 4;     // half-wave: supplies K = 4s+2h, 4s+2h+1

  const int gw = blockIdx.x * 8 + wave;   // 65536 waves total
  const int yt = gw & 31;                 // 32 y-tiles of 16
  const int xg = (gw >> 5) & 127;         // 128 x-groups of 4
  const int b  = gw >> 12;                // 16 batches

  const int y  = yt * 16 + l;
  const int yp = (y + 1) & (YD - 1);

  // preload this lane's 34 A-matrix coefficients (LDS, conflict-free)
  float ac0[17], ac1[17];
#pragma unroll
  for (int s = 0; s < 17; ++s) {
    ac0[s] = Bmat[(4*s + 2*h    )*16 + l];
    ac1[s] = Bmat[(4*s + 2*h + 1)*16 + l];
  }

  for (int it = 0; it < 4; ++it) {
    const int x  = xg * 4 + it;
    const int xp = (x + 1) & (XD - 1);

    const size_t pb = (size_t)b * CCH * XD * YD;
    const float2 W0  = plaqs[pb + ((size_t)0*XD + x )*YD + y ];
    const float2 W1  = plaqs[pb + ((size_t)1*XD + x )*YD + y ];
    const float2 lx  = links[pb + ((size_t)0*XD + x )*YD + y ];
    const float2 ly  = links[pb + ((size_t)1*XD + x )*YD + y ];
    const float2 W0x = plaqs[pb + ((size_t)0*XD + xp)*YD + y ];
    const float2 W1x = plaqs[pb + ((size_t)1*XD + xp)*YD + y ];
    const float2 W0y = plaqs[pb + ((size_t)0*XD + x )*YD + yp];
    const float2 W1y = plaqs[pb + ((size_t)1*XD + x )*YD + yp];

    // T_k: [W0, W1, lx*W0(x+1), lx*W1(x+1), ly*W0(y+1), ly*W1(y+1)]
    float Tre[6], Tim[6];
    Tre[0] = W0.x; Tim[0] = W0.y;
    Tre[1] = W1.x; Tim[1] = W1.y;
    { const float2 t = cmulc(lx, W0x); Tre[2] = t.x; Tim[2] = t.y; }
    { const float2 t = cmulc(lx, W1x); Tre[3] = t.x; Tim[3] = t.y; }
    { const float2 t = cmulc(ly, W0y); Tre[4] = t.x; Tim[4] = t.y; }
    { const float2 t = cmulc(ly, W1y); Tre[5] = t.x; Tim[5] = t.y; }

    const float ws0 = h ? W0.y : W0.x;   // half 0: ReW_c products, half 1: ImW_c products
    const float ws1 = h ? W1.y : W1.x;

    v8f acc = {0.f, 0.f, 0.f, 0.f, 0.f, 0.f, 0.f, 0.f};

#define WSTEP(s, b0v, b1v)                                                     \
    do {                                                                       \
      v2f av; av.x = ac0[s]; av.y = ac1[s];                                    \
      v2f bv; bv.x = (b0v);  bv.y = (b1v);                                     \
      acc = __builtin_amdgcn_wmma_f32_16x16x4_f32(                             \
          false, av, false, bv, (short)0, acc, false, false);                  \
    } while (0)

    WSTEP(0,  ws0 * Tre[0], ws0 * Tim[0]);
    WSTEP(1,  ws0 * Tre[1], ws0 * Tim[1]);
    WSTEP(2,  ws0 * Tre[2], ws0 * Tim[2]);
    WSTEP(3,  ws0 * Tre[3], ws0 * Tim[3]);
    WSTEP(4,  ws0 * Tre[4], ws0 * Tim[4]);
    WSTEP(5,  ws0 * Tre[5], ws0 * Tim[5]);
    WSTEP(6,  ws1 * Tre[0], ws1 * Tim[0]);
    WSTEP(7,  ws1 * Tre[1], ws1 * Tim[1]);
    WSTEP(8,  ws1 * Tre[2], ws1 * Tim[2]);
    WSTEP(9,  ws1 * Tre[3], ws1 * Tim[3]);
    WSTEP(10, ws1 * Tre[4], ws1 * Tim[4]);
    WSTEP(11, ws1 * Tre[5], ws1 * Tim[5]);
    WSTEP(12, h ? W1.x   : W0.x,   h ? W1.y   : W0.y);   // ReW0,ImW0 | ReW1,ImW1
    WSTEP(13, h ? Tre[1] : Tre[0], h ? Tim[1] : Tim[0]); // T0 | T1
    WSTEP(14, h ? Tre[3] : Tre[2], h ? Tim[3] : Tim[2]); // T2 | T3
    WSTEP(15, h ? Tre[5] : Tre[4], h ? Tim[5] : Tim[4]); // T4 | T5
    WSTEP(16, h ? 0.f : 1.f, 0.f);                       // const | padding
#undef WSTEP

    // D: VGPR r holds Out[M = r + 8h, N = l]; M = 2i+t  ->  lane stores (re,im) pairs
    const size_t ob = (size_t)b * NOUT * XD * YD;
#pragma unroll
    for (int q = 0; q < 4; ++q) {
      const int i = q + 4 * h;
      float2 v; v.x = acc[2*q]; v.y = acc[2*q + 1];
      out[ob + ((size_t)i * XD + x) * YD + y] = v;
    }
  }
}

extern "C" void kernel_launch(void* const* d_in, const int* in_sizes, int n_in,
                              void* d_out, int out_size, void* d_ws, size_t ws_size,
                              hipStream_t stream) {
  (void)in_sizes; (void)n_in; (void)out_size; (void)d_ws; (void)ws_size;
  const float2* plaqs = (const float2*)d_in[0];
  const float2* links = (const float2*)d_in[1];
  const float2* alpha = (const float2*)d_in[2];
  float2* out = (float2*)d_out;

  // 65536 waves = 16 b * 128 xg * 32 yt; 8 waves per 256-thread block
  dim3 grid(8192), block(256);
  hipLaunchKernelGGL(lconv_wmma_kernel, grid, block, 0, stream,
                     plaqs, links, alpha, out);
}